// Clopath_11562051961235
// MI455X (gfx1250) — compile-verified
//
#include <hip/hip_runtime.h>

typedef __attribute__((ext_vector_type(2))) float v2f;
typedef __attribute__((ext_vector_type(8))) float v8f;

// Problem constants (match reference)
constexpr int   D_  = 16;
constexpr int   B_  = 8;
constexpr int   N_  = 2048;
constexpr float ALPHA_X = 0.95f, ALPHA_P = 0.9f, ALPHA_D = 0.8f;
constexpr float WMAX = 2.0f;

constexpr int TPW     = 16;                // 16-wide o-tiles per wave
constexpr int OCHUNKS = N_ / (16 * TPW);   // 8

// One wave (32 threads) per workgroup.
// blockIdx.y = e (0..N-1), blockIdx.x = o-chunk (0..OCHUNKS-1).
//
// Per e:  B-matrix (4x16 f32 per WMMA step): col 0..7 = xbar_pre[d, col, e],
//         col 8..15 = Xd[d, col-8, e].  K index d = 4k + 2*(lane/16) + j.
// Per o-tile: A-matrix = dmap[d, e, o0+lane%16] (M = o, K = d), 4 chained
//         v_wmma_f32_16x16x4_f32 accumulate S_pot (cols 0..7) and S_dep
//         (cols 8..15) for the whole 16-o x 8-b tile.
__global__ __launch_bounds__(32) void clopath_wmma_kernel(
    const float* __restrict__ Xd,        // (D,B,N)
    const float* __restrict__ Xpost,     // (B,N)
    const float* __restrict__ u_pot,     // (B,N)
    const float* __restrict__ u_dep,     // (B,N)
    const float* __restrict__ xbar_pre,  // (D,B,N)
    const float* __restrict__ W,         // (B,N,N)
    const float* __restrict__ A_p,       // (N,N)
    const float* __restrict__ A_d,       // (N,N)
    const float* __restrict__ dmap,      // (D,N,N)
    float* __restrict__ out,             // (B,N,N)  = W copy
    float* __restrict__ W_new)           // (B,N,N)
{
    __shared__ float T[16 * 17];         // padded 16x16 transpose tile

    const int lane = threadIdx.x;        // 0..31
    const int col  = lane & 15;          // 0..15
    const int half = lane >> 4;          // 0 or 1
    const int e    = blockIdx.y;
    const int obase = blockIdx.x * (16 * TPW);

    // ---- Gather B matrix for this e (reused for all o-tiles) ----
    // breg[k] component j holds Bmat[d = 4k + 2*half + j][col].
    const float* bsrc = (col < 8)
        ? (xbar_pre + (size_t)col * N_ + e)
        : (Xd       + (size_t)(col - 8) * N_ + e);
    v2f breg[4];
#pragma unroll
    for (int k = 0; k < 4; ++k) {
        const int d0 = 4 * k + 2 * half;
        breg[k].x = bsrc[(size_t)d0       * ((size_t)B_ * N_)];
        breg[k].y = bsrc[(size_t)(d0 + 1) * ((size_t)B_ * N_)];
    }

    const size_t NN = (size_t)N_ * N_;

    for (int t = 0; t < TPW; ++t) {
        const int o0 = obase + t * 16;

        // ---- A tile: dmap[d, e, o0 + col]; coalesced 64B rows ----
        const float* asrc = dmap + (size_t)e * N_ + o0 + col;
        v8f acc = {};
#pragma unroll
        for (int k = 0; k < 4; ++k) {
            const int d0 = 4 * k + 2 * half;
            v2f a;
            a.x = asrc[(size_t)d0       * NN];
            a.y = asrc[(size_t)(d0 + 1) * NN];
            // D = A x B + C :  S[o][col] += sum_d dmap[d,e,o] * trace[d,b,e]
            acc = __builtin_amdgcn_wmma_f32_16x16x4_f32(
                false, a, false, breg[k], (short)0, acc, false, false);
        }

        // ---- Transpose through LDS: acc[g] = D[M = g + 8*half][col] ----
#pragma unroll
        for (int g = 0; g < 8; ++g)
            T[(g + 8 * half) * 17 + col] = acc[g];
        __syncthreads();   // single-wave WG: just orders DS traffic

        // ---- Epilogue: coalesced W / out / W_new, o along lanes ----
#pragma unroll
        for (int p = 0; p < 4; ++p) {
            const int   b    = 2 * p + half;       // 0..7
            const int   o    = o0 + col;           // o-contiguous across lanes
            const float spot = T[col * 17 + b];
            const float sdep = T[col * 17 + 8 + b];

            const size_t i_bo = (size_t)b * N_ + o;
            const size_t i_eo = (size_t)e * N_ + o;
            const size_t i_w  = (size_t)b * NN + i_eo;

            const float gp = Xpost[i_bo] * fmaxf(u_pot[i_bo], 0.0f);
            const float gd = fmaxf(u_dep[i_bo], 0.0f);
            const float w  = W[i_w];

            float wn = w + spot * A_p[i_eo] * gp - sdep * A_d[i_eo] * gd;
            wn = fminf(fmaxf(wn, 0.0f), WMAX);

            out[i_w]   = w;    // output #1: W before update
            W_new[i_w] = wn;   // output #2
        }
        __syncthreads();
    }
}

// Small elementwise trace filters (outputs #3, #4, #5).
__global__ __launch_bounds__(256) void trace_update_kernel(
    const float* __restrict__ Xd, const float* __restrict__ Vpost,
    const float* __restrict__ xbar_pre, const float* __restrict__ u_pot,
    const float* __restrict__ u_dep,
    float* __restrict__ xbar_new, float* __restrict__ u_pot_new,
    float* __restrict__ u_dep_new, int n_big, int n_small)
{
    const int i = blockIdx.x * blockDim.x + threadIdx.x;
    if (i < n_big)
        xbar_new[i] = ALPHA_X * xbar_pre[i] + (1.0f - ALPHA_X) * Xd[i];
    if (i < n_small) {
        u_pot_new[i] = ALPHA_P * u_pot[i] + (1.0f - ALPHA_P) * Vpost[i];
        u_dep_new[i] = ALPHA_D * u_dep[i] + (1.0f - ALPHA_D) * Vpost[i];
    }
}

extern "C" void kernel_launch(void* const* d_in, const int* in_sizes, int n_in,
                              void* d_out, int out_size, void* d_ws, size_t ws_size,
                              hipStream_t stream) {
    // setup_inputs() order:
    const float* Xd       = (const float*)d_in[0];  // (D,B,N)
    const float* Xpost    = (const float*)d_in[1];  // (B,N)
    const float* Vpost    = (const float*)d_in[2];  // (B,N)
    const float* xbar_pre = (const float*)d_in[3];  // (D,B,N)
    const float* u_pot    = (const float*)d_in[4];  // (B,N)
    const float* u_dep    = (const float*)d_in[5];  // (B,N)
    const float* W        = (const float*)d_in[6];  // (B,N,N)
    const float* A_p      = (const float*)d_in[7];  // (N,N)
    const float* A_d      = (const float*)d_in[8];  // (N,N)
    const float* dmap     = (const float*)d_in[9];  // (D,N,N)

    float* outp = (float*)d_out;
    const size_t BNN = (size_t)B_ * N_ * N_;
    const size_t DBN = (size_t)D_ * B_ * N_;
    const size_t BN  = (size_t)B_ * N_;

    float* o_W        = outp;                    // W (pre-update)
    float* o_Wnew     = outp + BNN;              // W_new
    float* o_xbar     = outp + 2 * BNN;          // xbar_pre_new
    float* o_upot     = outp + 2 * BNN + DBN;    // u_pot_new
    float* o_udep     = outp + 2 * BNN + DBN + BN;

    dim3 grid(OCHUNKS, N_);
    clopath_wmma_kernel<<<grid, 32, 0, stream>>>(
        Xd, Xpost, u_pot, u_dep, xbar_pre, W, A_p, A_d, dmap, o_W, o_Wnew);

    const int n_big = (int)DBN, n_small = (int)BN;
    trace_update_kernel<<<(n_big + 255) / 256, 256, 0, stream>>>(
        Xd, Vpost, xbar_pre, u_pot, u_dep, o_xbar, o_upot, o_udep,
        n_big, n_small);
}